// SymmetricCrossEntropy_88347477279174
// MI455X (gfx1250) — compile-verified
//
#include <hip/hip_runtime.h>
#include <math.h>

// ---------------- problem constants (from reference) ----------------
#define N_ROWS 2048
#define C_COLS 50257
#define ALPHA_F 0.1f
#define BETA_F 1.0f
#define PROB_MIN_F 1e-7f
#define PROB_MAX_F (1.0f - 1e-7f)
#define NEG_LOG_EPS_F 9.2103404f // -log(1e-4) as float32

// ---------------- kernel config ----------------
#define BS 512                       // threads per block (16 wave32)
#define NWAVES (BS / 32)
#define NF4 (C_COLS / 4)             // 12564 full float4 per row
#define TAIL (C_COLS & 3)            // 1 leftover element
#define NT ((NF4 + BS - 1) / BS)     // 25 tiles of BS float4
#define DEPTH 8                      // async pipeline depth (power of 2)
                                     // -> 64 KB in flight per WGP (1 block/WGP,
                                     //    VGPR-limited), enough to cover HBM
                                     //    latency at the per-WGP BW share

static_assert((DEPTH & (DEPTH - 1)) == 0, "DEPTH must be power of two");
static_assert(NT > DEPTH, "pipeline assumes NT > DEPTH");
static_assert(NT * BS >= NF4, "tiles must cover row");

// CDNA5 async global->LDS: GVS mode (SGPR base + VGPR byte offset).
// vdst = VGPR holding LDS byte address, vaddr = VGPR 32-bit byte offset.
__device__ __forceinline__ void async_tile_b128(const float* __restrict__ base,
                                                unsigned g_f4,
                                                const float4* lds_slot) {
  unsigned lds_addr = (unsigned)(size_t)lds_slot; // low 32 bits == LDS offset
  unsigned off = g_f4 * 16u;
  asm volatile("global_load_async_to_lds_b128 %0, %1, %2 offset:0"
               :
               : "v"(lds_addr), "v"(off), "s"(base)
               : "memory");
}

template <bool IS_MAX>
__device__ __forceinline__ float block_reduce(float v, volatile float* red) {
#pragma unroll
  for (int o = 16; o >= 1; o >>= 1) {
    float t = __shfl_xor(v, o, 32);
    v = IS_MAX ? fmaxf(v, t) : (v + t);
  }
  const int wid = threadIdx.x >> 5;
  const int lane = threadIdx.x & 31;
  if (lane == 0) red[wid] = v;
  __syncthreads();
  if (wid == 0) {
    float x = (lane < NWAVES) ? red[lane] : (IS_MAX ? -__builtin_inff() : 0.0f);
#pragma unroll
    for (int o = NWAVES / 2; o >= 1; o >>= 1) {
      float t = __shfl_xor(x, o, 32);
      x = IS_MAX ? fmaxf(x, t) : (x + t);
    }
    if (lane == 0) red[0] = x;
  }
  __syncthreads();
  float r = red[0];
  __syncthreads(); // red may be reused by the next reduction
  return r;
}

__device__ __forceinline__ float clampp(float p) {
  return fminf(fmaxf(p, PROB_MIN_F), PROB_MAX_F);
}

__global__ __launch_bounds__(BS, 1) void sce_row_kernel(
    const float* __restrict__ pred, const int* __restrict__ labels,
    float* __restrict__ row_loss) {
  __shared__ float4 stage[DEPTH][BS]; // 64 KB streaming buffers
  __shared__ float4 scrap[BS];        // sink for past-the-end pipeline issues
  __shared__ float red[NWAVES];

  const int row = blockIdx.x;
  const unsigned tid = threadIdx.x;
  const float* __restrict__ base = pred + (size_t)row * (size_t)C_COLS;

  // label logit (uniform per block; issue early to hide latency)
  const int lab = labels[row];
  const float x_label = base[lab];

  // tail element(s): C % 4 leftover scalars
  const bool tail_v = (TAIL > 0) && (tid < (unsigned)TAIL);
  const float xtail = tail_v ? base[NF4 * 4 + tid] : -__builtin_inff();

  // ---------------- async global -> LDS -> registers pipeline ----------------
#pragma unroll
  for (int kk = 0; kk < DEPTH; ++kk) {
    unsigned g = (unsigned)(kk * BS) + tid;
    if (g >= (unsigned)NF4) g = (unsigned)(NF4 - 1); // clamp: stay inside row
    async_tile_b128(base, g, &stage[kk][tid]);
  }

  float4 vals[NT];
#pragma unroll
  for (int k = 0; k < NT; ++k) {
    // DEPTH tiles outstanding -> waiting for <=DEPTH-1 completes tile k
    asm volatile("s_wait_asynccnt 0x7" ::: "memory");
    vals[k] = stage[k & (DEPTH - 1)][tid];
    // make sure our LDS read retired before this buffer is re-targeted
    asm volatile("s_wait_dscnt 0x0" ::: "memory");
    const int kk = k + DEPTH;
    unsigned g = (unsigned)(kk * BS) + tid;
    if (g >= (unsigned)NF4) g = (unsigned)(NF4 - 1);
    const float4* dst = (kk < NT) ? &stage[kk & (DEPTH - 1)][tid] : &scrap[tid];
    async_tile_b128(base, g, dst); // every wave issues every iter: uniform cnt
  }

  // lanes whose last tile chunk is past the row: neutralize
  const bool last_valid = ((unsigned)((NT - 1) * BS) + tid) < (unsigned)NF4;
  if (!last_valid) {
    const float ninf = -__builtin_inff();
    vals[NT - 1] = make_float4(ninf, ninf, ninf, ninf);
  }

  // ---------------- phase A: row max ----------------
  float m = xtail;
#pragma unroll
  for (int k = 0; k < NT; ++k) {
    float4 v = vals[k];
    m = fmaxf(m, fmaxf(fmaxf(v.x, v.y), fmaxf(v.z, v.w)));
  }
  const float M = block_reduce<true>(m, red);

  // ---------------- phase B: e = exp(x - M), S = sum(e) ----------------
  float ssum = 0.0f;
#pragma unroll
  for (int k = 0; k < NT; ++k) {
    float4 v = vals[k];
    float4 e;
    e.x = __expf(v.x - M);
    e.y = __expf(v.y - M);
    e.z = __expf(v.z - M);
    e.w = __expf(v.w - M);
    vals[k] = e;
    ssum += (e.x + e.y) + (e.z + e.w);
  }
  const float etail = __expf(xtail - M); // 0 for invalid lanes (x = -inf)
  ssum += etail;
  const float S = block_reduce<false>(ssum, red);
  const float invS = 1.0f / S;

  // ---------------- phase C: sum of clamped probabilities ----------------
  float cs = 0.0f;
#pragma unroll
  for (int k = 0; k < NT; ++k) {
    float4 e = vals[k];
    float t = clampp(e.x * invS) + clampp(e.y * invS) + clampp(e.z * invS) +
              clampp(e.w * invS);
    if (k == NT - 1) t = last_valid ? t : 0.0f; // clamp would lift 0 -> 1e-7
    cs += t;
  }
  cs += tail_v ? clampp(etail * invS) : 0.0f;
  const float CS = block_reduce<false>(cs, red);

  if (tid == 0) {
    const float logZ = M + logf(S);
    const float ce = logZ - x_label;
    const float p_lab = clampp(__expf(x_label - M) * invS);
    const float rce = NEG_LOG_EPS_F * (CS - p_lab);
    row_loss[row] = (ALPHA_F * ce + BETA_F * rce) * (1.0f / (float)N_ROWS);
  }
}

// deterministic final reduction of 2048 per-row contributions
__global__ __launch_bounds__(256) void sce_final_reduce(
    const float* __restrict__ part, float* __restrict__ out) {
  __shared__ float red[8];
  float s = 0.0f;
  for (int i = threadIdx.x; i < N_ROWS; i += 256) s += part[i];
#pragma unroll
  for (int o = 16; o >= 1; o >>= 1) s += __shfl_xor(s, o, 32);
  const int wid = threadIdx.x >> 5;
  const int lane = threadIdx.x & 31;
  if (lane == 0) red[wid] = s;
  __syncthreads();
  if (threadIdx.x == 0) {
    float t = 0.0f;
#pragma unroll
    for (int w = 0; w < 8; ++w) t += red[w];
    out[0] = t;
  }
}

extern "C" void kernel_launch(void* const* d_in, const int* in_sizes, int n_in,
                              void* d_out, int out_size, void* d_ws,
                              size_t ws_size, hipStream_t stream) {
  (void)in_sizes;
  (void)n_in;
  (void)out_size;
  (void)ws_size;
  const float* pred = (const float*)d_in[0];
  const int* labels = (const int*)d_in[1];
  float* row_loss = (float*)d_ws; // 2048 floats of scratch
  sce_row_kernel<<<N_ROWS, BS, 0, stream>>>(pred, labels, row_loss);
  sce_final_reduce<<<1, 256, 0, stream>>>(row_loss, (float*)d_out);
}